// CARAFE_3_33947421508453
// MI455X (gfx1250) — compile-verified
//
#include <hip/hip_runtime.h>
#include <hip/hip_bf16.h>
#include <math.h>

// ---------------------------------------------------------------------------
// CARAFE on gfx1250 (MI455X, wave32):
//  - conv1 (1x1, 256->64) + conv2 (3x3, 64->100) as V_WMMA_F32_16X16X4_F32
//    GEMMs, one 16x16 f32 tile per wave.
//  - pixel-shuffle + softmax.
//  - reassembly: bilinear-upsample + 5x5 dilated(2) dynamic filter collapsed
//    algebraically into a 6x6 stencil on X (36 FMA/output instead of ~175).
// Op is HBM-bound (~270 MB @ 23.3 TB/s ~= 12us); intermediates L2-resident.
// ---------------------------------------------------------------------------

typedef float v2f __attribute__((ext_vector_type(2)));
typedef float v8f __attribute__((ext_vector_type(8)));

#define C_IN   256
#define C_MID  64
#define C_ENC  100     // (SCALE*K_UP)^2
#define H_IN   64
#define W_IN   64
#define HW_IN  4096
#define H_OUT  128
#define W_OUT  128
#define HW_OUT 16384
#define BATCH  8
#define KW     25      // K_UP^2
#define EPSV   1e-5f

#if __has_builtin(__builtin_amdgcn_global_load_async_to_lds_b32) && \
    __has_builtin(__builtin_amdgcn_s_wait_asynccnt)
#define USE_ASYNC_LDS 1
typedef __attribute__((address_space(1))) int* as1_i32p;
typedef __attribute__((address_space(3))) int* as3_i32p;
#endif

__device__ __forceinline__ int clampi(int v, int lo, int hi) {
  return v < lo ? lo : (v > hi ? hi : v);
}

// ---------------------------------------------------------------------------
// Kernel 1: 1x1 conv GEMM  Y1[b,co,p] = relu(bn1(sum_ci W[co,ci]*X[b,ci,p]))
// M=64 (co), K=256 (ci), N=4096/batch. One wave = 16x16 tile, 64 WMMAs.
// ---------------------------------------------------------------------------
__global__ void __launch_bounds__(256) carafe_conv1_wmma(
    const float* __restrict__ X,  const float* __restrict__ Wc,
    const float* __restrict__ Bc, const float* __restrict__ g,
    const float* __restrict__ be, const float* __restrict__ mu,
    const float* __restrict__ va, float* __restrict__ Y1)
{
  const int wave = blockIdx.x * (blockDim.x >> 5) + (threadIdx.x >> 5);
  const int coT  = wave & 3;            // 4 tiles of co
  const int pT   = (wave >> 2) & 255;   // 256 position tiles
  const int b    = wave >> 10;          // batch
  const int lane = threadIdx.x & 31;
  const int half = lane >> 4;           // K-pair selector (ISA 16x4 A layout)
  const int lid  = lane & 15;
  const int co0  = coT << 4;
  const int p0   = pT << 4;

  const float* Xb   = X + (size_t)b * C_IN * HW_IN;
  const float* Arow = Wc + (size_t)(co0 + lid) * C_IN;   // contiguous in K

  v8f acc = {0.f,0.f,0.f,0.f,0.f,0.f,0.f,0.f};
  for (int k = 0; k < C_IN; k += 4) {
    const int k0 = k + half * 2;
    v2f a;  a.x = Arow[k0];  a.y = Arow[k0 + 1];          // merges to b64
    const float* xc = Xb + (size_t)k0 * HW_IN + (p0 + lid);
    v2f bv; bv.x = xc[0];   bv.y = xc[HW_IN];             // coalesced
    acc = __builtin_amdgcn_wmma_f32_16x16x4_f32(false, a, false, bv,
                                                (short)0, acc, false, false);
  }
  #pragma unroll
  for (int r = 0; r < 8; ++r) {          // D: VGPR r -> M=r+half*8, N=lid
    const int co = co0 + r + half * 8;
    float y = acc[r] + Bc[co];
    y = g[co] * (y - mu[co]) * rsqrtf(va[co] + EPSV) + be[co];
    Y1[((size_t)b * C_MID + co) * HW_IN + p0 + lid] = fmaxf(y, 0.f);
  }
}

// ---------------------------------------------------------------------------
// Kernel 2: 3x3 conv as GEMM with taps hoisted: for each (ky,kx) accumulate a
// K=64 (ci) GEMM. Inner body = 4 loads + 2 cndmask + 1 WMMA. M=100 padded to
// 112: A-row pointer clamped to row 99 (rows >=100 never stored, D-row m only
// depends on A-row m). Borders: clamped address + cndmask-to-zero.
// ---------------------------------------------------------------------------
__global__ void __launch_bounds__(256) carafe_conv2_wmma(
    const float* __restrict__ Y1, const float* __restrict__ We,
    const float* __restrict__ Be, const float* __restrict__ g,
    const float* __restrict__ be, const float* __restrict__ mu,
    const float* __restrict__ va, float* __restrict__ Y2)
{
  const int wave = blockIdx.x * (blockDim.x >> 5) + (threadIdx.x >> 5);
  const int coT  = wave % 7;             // 7 tiles cover M=100..112
  const int t    = wave / 7;
  const int pT   = t & 255;
  const int b    = t >> 8;
  const int lane = threadIdx.x & 31;
  const int half = lane >> 4;
  const int lid  = lane & 15;
  const int co0  = coT << 4;
  const int p0   = pT << 4;
  const int yo   = p0 >> 6;              // output row
  const int x0   = p0 & 63;

  const int coA = clampi(co0 + lid, 0, C_ENC - 1);
  const float* Abase = We + (size_t)coA * 576 + (half * 2) * 9;  // + ci*9 later
  const float* Yb = Y1 + (size_t)b * C_MID * HW_IN + ((size_t)(half * 2) << 12);
  const int xn = x0 + lid;

  v8f acc = {0.f,0.f,0.f,0.f,0.f,0.f,0.f,0.f};
  for (int ky = 0; ky < 3; ++ky) {
    const int yy = yo + ky - 1;
    if (yy < 0 || yy >= H_IN) continue;                 // wave-uniform skip
    #pragma unroll
    for (int kx = 0; kx < 3; ++kx) {
      const int xx  = xn + kx - 1;
      const bool ok = (unsigned)xx < (unsigned)W_IN;    // per-lane, hoisted
      const float* ap = Abase + ky * 3 + kx;
      const float* bp = Yb + (yy << 6) + clampi(xx, 0, W_IN - 1);
      #pragma unroll 4
      for (int k = 0; k < C_MID; k += 4) {
        v2f a, bv;
        a.x = ap[0];                                    // stride 9 in ci
        a.y = ap[9];
        const float v0 = bp[0];
        const float v1 = bp[HW_IN];
        bv.x = ok ? v0 : 0.f;
        bv.y = ok ? v1 : 0.f;
        acc = __builtin_amdgcn_wmma_f32_16x16x4_f32(false, a, false, bv,
                                                    (short)0, acc, false, false);
        ap += 36;                                       // 4 ci * 9
        bp += 4 * HW_IN;
      }
    }
  }
  #pragma unroll
  for (int r = 0; r < 8; ++r) {
    const int co = co0 + r + half * 8;
    if (co < C_ENC) {
      float y = acc[r] + Be[co];
      y = g[co] * (y - mu[co]) * rsqrtf(va[co] + EPSV) + be[co];
      Y2[((size_t)b * C_ENC + co) * HW_IN + p0 + lid] = y;
    }
  }
}

// ---------------------------------------------------------------------------
// Kernel 3: pixel-shuffle(r=2) + softmax over the 25 reassembly weights.
// ---------------------------------------------------------------------------
__global__ void __launch_bounds__(256) carafe_softmax(
    const float* __restrict__ Y2, float* __restrict__ Wb)
{
  const int idx = blockIdx.x * blockDim.x + threadIdx.x;  // 8*128*128
  const int w = idx & 127, h = (idx >> 7) & 127, b = idx >> 14;
  const int y = h >> 1, x = w >> 1;
  const int sub = ((h & 1) << 1) | (w & 1);               // channel % 4
  const float* base = Y2 + ((size_t)b * C_ENC + sub) * HW_IN + (y << 6) + x;

  float v[KW], mx = -INFINITY;
  #pragma unroll
  for (int k = 0; k < KW; ++k) { v[k] = base[(size_t)k * 4 * HW_IN]; mx = fmaxf(mx, v[k]); }
  float s = 0.f;
  #pragma unroll
  for (int k = 0; k < KW; ++k) { v[k] = __expf(v[k] - mx); s += v[k]; }
  const float inv = 1.f / s;
  float* o = Wb + (size_t)b * KW * HW_OUT + (h << 7) + w;
  #pragma unroll
  for (int k = 0; k < KW; ++k) o[(size_t)k * HW_OUT] = v[k] * inv;
}

// ---------------------------------------------------------------------------
// Kernel 4: collapsed reassembly. Taps step by 2 in output space = 1 in input
// space, and share one bilinear fraction per axis (parity-determined). So
// upsample+pad+25-tap filter == 6x6 stencil on X with per-pixel weights
//   G[l][m] = sum_{dy,dx in {0,1}} ay[dy]*ax[dx]*WM[l-dy][m-dx]
// (WM = validity-masked softmax weights; edge clamp folded into index tables).
// 36 loads + 36 FMAs per channel; 2-channel unroll -> dual-issue FMA chains.
// Softmax weights staged via async global->LDS when the toolchain has it.
// ---------------------------------------------------------------------------
__global__ void __launch_bounds__(256) carafe_reassemble(
    const float* __restrict__ X, const float* __restrict__ Wb,
    float* __restrict__ out)
{
  const int idx = blockIdx.x * blockDim.x + threadIdx.x;  // 8*128*128
  const int w = idx & 127, h = (idx >> 7) & 127, b = idx >> 14;
  const float* wp = Wb + (size_t)b * KW * HW_OUT + (h << 7) + w;

#ifdef USE_ASYNC_LDS
  __shared__ float sW[KW * 256];
  {
    const int tid = threadIdx.x;
    #pragma unroll
    for (int k = 0; k < KW; ++k) {
      __builtin_amdgcn_global_load_async_to_lds_b32(
          (as1_i32p)(wp + (size_t)k * HW_OUT),
          (as3_i32p)(&sW[k * 256 + tid]),
          0, 0);
    }
    __builtin_amdgcn_s_wait_asynccnt(0);   // own data only; no barrier needed
  }
#endif

  // validity masks (zero padding of the upsampled field)
  bool vy[5], vx[5];
  #pragma unroll
  for (int i = 0; i < 5; ++i) {
    vy[i] = (unsigned)(h - 4 + 2 * i) < (unsigned)H_OUT;
    vx[i] = (unsigned)(w - 4 + 2 * i) < (unsigned)W_OUT;
  }
  float WM[5][5];
  #pragma unroll
  for (int i = 0; i < 5; ++i)
    #pragma unroll
    for (int j = 0; j < 5; ++j) {
#ifdef USE_ASYNC_LDS
      const float wv = sW[(i * 5 + j) * 256 + (int)threadIdx.x];
#else
      const float wv = wp[(size_t)(i * 5 + j) * HW_OUT];
#endif
      WM[i][j] = (vy[i] && vx[j]) ? wv : 0.f;
    }

  // bilinear structure: src = q*0.5-0.25 -> frac 0.75 (even) / 0.25 (odd)
  const int   tt = h >> 1, ss = w >> 1;
  const float a1 = (h & 1) ? 0.25f : 0.75f, a0 = 1.f - a1;   // weight X[f+1], X[f]
  const float b1 = (w & 1) ? 0.25f : 0.75f, b0 = 1.f - b1;
  const int   r0 = (h & 1) ? (tt - 2) : (tt - 3);
  const int   c0 = (w & 1) ? (ss - 2) : (ss - 3);

  float G[6][6];
  #pragma unroll
  for (int l = 0; l < 6; ++l)
    #pragma unroll
    for (int m = 0; m < 6; ++m) {
      float gv = 0.f;
      if (l < 5 && m < 5) gv += a0 * b0 * WM[l][m];
      if (l < 5 && m > 0) gv += a0 * b1 * WM[l][m - 1];
      if (l > 0 && m < 5) gv += a1 * b0 * WM[l - 1][m];
      if (l > 0 && m > 0) gv += a1 * b1 * WM[l - 1][m - 1];
      G[l][m] = gv;
    }

  int R[6], C[6];   // clamped (resize edge handling), all loads in-bounds
  #pragma unroll
  for (int l = 0; l < 6; ++l) {
    R[l] = clampi(r0 + l, 0, H_IN - 1) << 6;
    C[l] = clampi(c0 + l, 0, W_IN - 1);
  }

  const float* Xb = X + (size_t)b * C_IN * HW_IN;
  float* ob = out + (size_t)b * C_IN * HW_OUT + (h << 7) + w;
  for (int c = 0; c < C_IN; c += 2) {
    const float* Xp0 = Xb + (size_t)c * HW_IN;
    const float* Xp1 = Xp0 + HW_IN;
    float acc0 = 0.f, acc1 = 0.f;      // two independent chains -> VOPD
    #pragma unroll
    for (int l = 0; l < 6; ++l) {
      const float* rp0 = Xp0 + R[l];
      const float* rp1 = Xp1 + R[l];
      #pragma unroll
      for (int m = 0; m < 6; ++m) {
        acc0 = fmaf(G[l][m], rp0[C[m]], acc0);
        acc1 = fmaf(G[l][m], rp1[C[m]], acc1);
      }
    }
    ob[(size_t)c * HW_OUT]       = acc0;
    ob[(size_t)(c + 1) * HW_OUT] = acc1;
  }
}

// ---------------------------------------------------------------------------
extern "C" void kernel_launch(void* const* d_in, const int* in_sizes, int n_in,
                              void* d_out, int out_size, void* d_ws, size_t ws_size,
                              hipStream_t stream) {
  const float* X      = (const float*)d_in[0];
  const float* comp_w = (const float*)d_in[1];
  const float* comp_b = (const float*)d_in[2];
  const float* bn1_g  = (const float*)d_in[3];
  const float* bn1_b  = (const float*)d_in[4];
  const float* bn1_m  = (const float*)d_in[5];
  const float* bn1_v  = (const float*)d_in[6];
  const float* enc_w  = (const float*)d_in[7];
  const float* enc_b  = (const float*)d_in[8];
  const float* bn2_g  = (const float*)d_in[9];
  const float* bn2_b  = (const float*)d_in[10];
  const float* bn2_m  = (const float*)d_in[11];
  const float* bn2_v  = (const float*)d_in[12];
  float* out = (float*)d_out;

  // workspace layout (33 MB total -> L2-resident on MI455X, 192 MB L2)
  float* Y1 = (float*)d_ws;                       // 8*64*4096   =  8 MiB
  float* Y2 = Y1 + (size_t)BATCH * C_MID * HW_IN; // 8*100*4096  = 12.5 MiB
  float* Wb = Y2 + (size_t)BATCH * C_ENC * HW_IN; // 8*25*16384  = 12.5 MiB

  carafe_conv1_wmma<<<1024, 256, 0, stream>>>(X, comp_w, comp_b,
      bn1_g, bn1_b, bn1_m, bn1_v, Y1);
  carafe_conv2_wmma<<<1792, 256, 0, stream>>>(Y1, enc_w, enc_b,
      bn2_g, bn2_b, bn2_m, bn2_v, Y2);
  carafe_softmax<<<512, 256, 0, stream>>>(Y2, Wb);
  carafe_reassemble<<<512, 256, 0, stream>>>(X, Wb, out);
}